// FIURIModule_50070728737564
// MI455X (gfx1250) — compile-verified
//
#include <hip/hip_runtime.h>
#include <math.h>

// ---------------------------------------------------------------------------
// FIURI forward (new_o only).  B=16 fixed by problem.
// Phase 1: transpose o_pre / in_state to (N,16) + zero gj accumulator.
// Phase 2: edge scatter-add, edge stream staged to LDS via gfx1250 async
//          loads (double buffered, ASYNCcnt-synchronized).
// Phase 3: elementwise finalize.
// ---------------------------------------------------------------------------

#define TILE 256

#if __has_builtin(__builtin_amdgcn_global_load_async_to_lds_b32)
#define HAVE_ASYNC 1
#else
#define HAVE_ASYNC 0
#endif

#if __has_builtin(__builtin_amdgcn_s_wait_asynccnt)
#define WAIT_ASYNC(n) __builtin_amdgcn_s_wait_asynccnt(n)
#elif HAVE_ASYNC
#define WAIT_ASYNC(n) asm volatile("s_wait_asynccnt %0" :: "i"(n) : "memory")
#else
#define WAIT_ASYNC(n) ((void)0)
#endif

typedef __attribute__((address_space(1))) int gint;   // global AS int
typedef __attribute__((address_space(3))) int lint;   // LDS AS int

__device__ __forceinline__ void async_copy_b32(const void* g, void* l) {
#if HAVE_ASYNC
    void* gg = const_cast<void*>(g);
    __builtin_amdgcn_global_load_async_to_lds_b32((gint*)gg, (lint*)l,
                                                  /*offset=*/0, /*cpol=*/0);
#else
    *(unsigned*)l = *(const unsigned*)g;
#endif
}

// --------------------------- phase 1 kernels -------------------------------

__global__ void fiuri_prep(const float* __restrict__ opre,
                           const float* __restrict__ instate,
                           float* __restrict__ opre_t,
                           float* __restrict__ instate_t,
                           float* __restrict__ gj_t, int N) {
    int i = blockIdx.x * blockDim.x + threadIdx.x;   // over N*16
    if (i < N * 16) {
        int b = i & 15;
        int n = i >> 4;
        opre_t[i]    = opre[b * N + n];
        instate_t[i] = instate[b * N + n];
        gj_t[i]      = 0.0f;
    }
}

__global__ void fiuri_zero(float* __restrict__ p, int n) {
    int i = blockIdx.x * blockDim.x + threadIdx.x;
    if (i < n) p[i] = 0.0f;
}

// --------------------------- phase 2 kernel --------------------------------
// One block stages TILE edges (src,dst,w) to LDS with async loads, double
// buffered across grid-stride chunks.  Thread t handles batch b = t&15 for
// 16 staged edges; lanes 0-15 of a wave read 16 contiguous floats of the
// transposed node arrays (single 64B run) and issue 16 contiguous atomics.

__global__ __launch_bounds__(TILE) void fiuri_edges(
    const int* __restrict__ gsrc, const int* __restrict__ gdst,
    const float* __restrict__ gw,
    const float* __restrict__ opre, const float* __restrict__ instate,
    float* __restrict__ accum, long long E, int N, int transposed)
{
    __shared__ int   s_src[2][TILE];
    __shared__ int   s_dst[2][TILE];
    __shared__ float s_w  [2][TILE];

    const int t  = threadIdx.x;
    const int b  = t & 15;
    const int er = t >> 4;                       // 0..15
    const long long nChunks = (E + TILE - 1) / TILE;
    const long long stride  = gridDim.x;
    long long c = blockIdx.x;
    int buf = 0;

    if (c < nChunks) {                           // prime buffer 0
        long long ei = c * TILE + t;
        if (ei >= E) ei = E - 1;                 // clamp: keep EXEC all-ones
        async_copy_b32(gsrc + ei, &s_src[0][t]);
        async_copy_b32(gdst + ei, &s_dst[0][t]);
        async_copy_b32(gw   + ei, &s_w  [0][t]);
    }

    for (; c < nChunks; c += stride) {
        long long cn = c + stride;
        if (cn < nChunks) {                      // prefetch next chunk
            long long ei = cn * TILE + t;
            if (ei >= E) ei = E - 1;
            int nb = buf ^ 1;
            async_copy_b32(gsrc + ei, &s_src[nb][t]);
            async_copy_b32(gdst + ei, &s_dst[nb][t]);
            async_copy_b32(gw   + ei, &s_w  [nb][t]);
            WAIT_ASYNC(3);                       // current chunk's 3 done
        } else {
            WAIT_ASYNC(0);
        }
        __syncthreads();

        const long long base = c * TILE;
#pragma unroll
        for (int k = 0; k < 16; ++k) {
            const int el = er + (k << 4);
            if (base + el < E) {
                const int   s  = s_src[buf][el];
                const int   d  = s_dst[buf][el];
                const float wv = s_w  [buf][el];
                float Oj, En; int ai;
                if (transposed) {
                    Oj = opre[(s << 4) + b];
                    En = instate[(d << 4) + b];
                    ai = (d << 4) + b;
                } else {
                    Oj = opre[b * N + s];
                    En = instate[b * N + d];
                    ai = b * N + d;
                }
                const float x = Oj * wv;
                atomicAdd(&accum[ai], (Oj >= En) ? x : -x);
            }
        }
        __syncthreads();                         // buffer reuse safety
        buf ^= 1;
    }
}

// --------------------------- phase 3 kernel --------------------------------

__global__ void fiuri_finalize(const float* __restrict__ ex,
                               const float* __restrict__ inr,
                               const float* __restrict__ instate,
                               const float* __restrict__ thr,
                               const float* __restrict__ accum,
                               float* __restrict__ out, int N, int transposed) {
    int n = blockIdx.x * blockDim.x + threadIdx.x;
    int b = blockIdx.y;
    if (n < N) {
        int i = b * N + n;
        float gj = transposed ? accum[(n << 4) + b] : accum[i];
        float S  = instate[i] + (ex[i] - inr[i]) + gj;
        S = fminf(10.0f, fmaxf(-10.0f, S));
        float x = thr[n];
        float T = fmaxf(x, 0.0f) + log1pf(expf(-fabsf(x)));  // stable softplus
        out[i] = fmaxf(S - T, 0.0f);
    }
}

// --------------------------- launcher --------------------------------------

extern "C" void kernel_launch(void* const* d_in, const int* in_sizes, int n_in,
                              void* d_out, int out_size, void* d_ws, size_t ws_size,
                              hipStream_t stream)
{
    const float* ex   = (const float*)d_in[0];
    const float* inr  = (const float*)d_in[1];
    const float* opre = (const float*)d_in[2];
    const float* inst = (const float*)d_in[3];
    const int*   src  = (const int*)d_in[4];
    const int*   dst  = (const int*)d_in[5];
    const float* w    = (const float*)d_in[6];
    const float* thr  = (const float*)d_in[7];
    // d_in[8] (decay) unused: reference returns new_o only.

    const int       N = in_sizes[7];
    const long long E = in_sizes[4];

    float* out = (float*)d_out;

    const long long nChunks = (E + TILE - 1) / TILE;
    int eblocks = (int)(nChunks < 4096 ? nChunks : 4096);
    if (eblocks < 1) eblocks = 1;

    dim3 fgrid((N + 255) / 256, 16);
    const size_t need = (size_t)N * 16 * sizeof(float) * 3;

    if (ws_size >= need) {
        float* opre_t = (float*)d_ws;
        float* inst_t = opre_t + (size_t)N * 16;
        float* gj_t   = inst_t + (size_t)N * 16;
        int pblocks = (N * 16 + 255) / 256;
        fiuri_prep<<<pblocks, 256, 0, stream>>>(opre, inst, opre_t, inst_t, gj_t, N);
        fiuri_edges<<<eblocks, TILE, 0, stream>>>(src, dst, w, opre_t, inst_t,
                                                  gj_t, E, N, 1);
        fiuri_finalize<<<fgrid, 256, 0, stream>>>(ex, inr, inst, thr, gj_t, out, N, 1);
    } else {
        int zblocks = (N * 16 + 255) / 256;
        fiuri_zero<<<zblocks, 256, 0, stream>>>(out, N * 16);
        fiuri_edges<<<eblocks, TILE, 0, stream>>>(src, dst, w, opre, inst,
                                                  out, E, N, 0);
        fiuri_finalize<<<fgrid, 256, 0, stream>>>(ex, inr, inst, thr, out, out, N, 0);
    }
}